// PyCDGM_90512140796748
// MI455X (gfx1250) — compile-verified
//
#include <hip/hip_runtime.h>

// ---------------------------------------------------------------------------
// MI455X (gfx1250, wave32). All GEMMs use V_WMMA_F32_16X16X4_F32.
// Strip-mined: one wave computes a 16 x (16*NSTRIP) strip of C, reusing the
// A fragment across NSTRIP WMMAs per K-step. All dims compile-time constants.
// ---------------------------------------------------------------------------

typedef float v8f __attribute__((ext_vector_type(8)));
typedef float v2f __attribute__((ext_vector_type(2)));

#define NN    2048
#define D_IN  128
#define HH    256
#define D_EMB 64

__device__ __forceinline__ float sigmoid_f(float z) {
    return 1.0f / (1.0f + __expf(-z));
}

// ---------------------------------------------------------------------------
// C[M,N] = epilogue(A[M,K] @ B[K,N] (+ bias))
//   ACT = 0: +bias            ACT = 1: relu(+bias)
//   ACT = 2: relu(val / deg[m])   (no bias)
// One wave per 16 x (16*NSTRIP) strip. Grid covers strips exactly.
// ---------------------------------------------------------------------------
template<int N, int K, int NSTRIP, int ACT>
__global__ void wmma_gemm_strip(const float* __restrict__ A,
                                const float* __restrict__ B,
                                const float* __restrict__ bias,  // ACT 0/1
                                const float* __restrict__ deg,   // ACT 2
                                float* __restrict__ C)
{
    static_assert(N % (16 * NSTRIP) == 0, "strip width must divide N");
    constexpr int STRIPW = 16 * NSTRIP;
    constexpr int SN = N / STRIPW;          // strips per 16-row band

    const int lane  = threadIdx.x & 31;
    const int wave  = threadIdx.x >> 5;
    const int strip = blockIdx.x * (blockDim.x >> 5) + wave;
    const int tm  = (strip / SN) * 16;
    const int tn0 = (strip % SN) * STRIPW;

    const int r = lane & 15;   // A-frag row / B-frag col within tile
    const int h = lane >> 4;   // lane-half: selects K pair (A/B), +8 rows (D)

    // ISA 7.12.2 fp32 16x4 A frag: lanes0-15 K={k,k+1}, lanes16-31 K={k+2,k+3}
    const float* __restrict__ arow  = A + (size_t)(tm + r) * K + h * 2;
    const float* __restrict__ bbase = B + tn0 + r + (size_t)(h * 2) * N;

    v8f acc[NSTRIP] = {};
    for (int k = 0; k < K; k += 4) {
        v2f a;
        a.x = arow[k];
        a.y = arow[k + 1];
        #pragma unroll
        for (int t = 0; t < NSTRIP; ++t) {
            v2f b;
            b.x = bbase[(size_t)k * N + t * 16];
            b.y = bbase[(size_t)(k + 1) * N + t * 16];
            acc[t] = __builtin_amdgcn_wmma_f32_16x16x4_f32(
                         false, a, false, b, (short)0, acc[t], false, false);
        }
    }

    // D layout: VGPR i -> M = tm + i + h*8 ; N = tn0 + t*16 + (lane&15)
    #pragma unroll
    for (int t = 0; t < NSTRIP; ++t) {
        const int n = tn0 + t * 16 + r;
        const float bv = (ACT == 2) ? 0.0f : bias[n];
        #pragma unroll
        for (int i = 0; i < 8; ++i) {
            const int m = tm + h * 8 + i;
            float v = acc[t][i] + bv;
            if (ACT == 1) v = fmaxf(v, 0.0f);
            if (ACT == 2) v = fmaxf(v / deg[m], 0.0f);
            C[(size_t)m * N + n] = v;
        }
    }
}

// ---------------------------------------------------------------------------
// Fused Gram + distance + sigmoid + identity (strip-mined, K = D_EMB):
//   G = emb @ emb^T
//   Aout[m,n] = sigmoid((1+temp) * -(sq[m]+sq[n]-2G) + (5+theta)) + (m==n)
// ---------------------------------------------------------------------------
template<int NSTRIP>
__global__ void gram_prob_strip(const float* __restrict__ emb,
                                const float* __restrict__ sq,
                                const float* __restrict__ tempP,
                                const float* __restrict__ thetaP,
                                float* __restrict__ Aout)
{
    constexpr int Nn = NN, Kd = D_EMB;
    constexpr int STRIPW = 16 * NSTRIP;
    constexpr int SN = Nn / STRIPW;

    const int lane  = threadIdx.x & 31;
    const int wave  = threadIdx.x >> 5;
    const int strip = blockIdx.x * (blockDim.x >> 5) + wave;
    const int tm  = (strip / SN) * 16;
    const int tn0 = (strip % SN) * STRIPW;

    const int r = lane & 15;
    const int h = lane >> 4;

    const float* __restrict__ arow = emb + (size_t)(tm + r) * Kd + h * 2;

    v8f acc[NSTRIP] = {};
    for (int k = 0; k < Kd; k += 4) {
        v2f a;
        a.x = arow[k];
        a.y = arow[k + 1];
        #pragma unroll
        for (int t = 0; t < NSTRIP; ++t) {
            // B = emb^T : B[k][n] = emb[n][k]
            const float* __restrict__ brow =
                emb + (size_t)(tn0 + t * 16 + r) * Kd + h * 2;
            v2f b;
            b.x = brow[k];
            b.y = brow[k + 1];
            acc[t] = __builtin_amdgcn_wmma_f32_16x16x4_f32(
                         false, a, false, b, (short)0, acc[t], false, false);
        }
    }

    const float t_scale = 1.0f + *tempP;
    const float t_bias  = 5.0f + *thetaP;
    #pragma unroll
    for (int t = 0; t < NSTRIP; ++t) {
        const int n = tn0 + t * 16 + r;
        const float sqn = sq[n];
        #pragma unroll
        for (int i = 0; i < 8; ++i) {
            const int m = tm + h * 8 + i;
            const float d = -(sq[m] + sqn - 2.0f * acc[t][i]);
            float p = sigmoid_f(t_scale * d + t_bias);
            if (m == n) p += 1.0f;
            Aout[(size_t)m * Nn + n] = p;
        }
    }
}

// sq[i] = sum_k emb[i,k]^2  (one thread per row)
__global__ void rowsq_f32(const float* __restrict__ emb,
                          float* __restrict__ sq)
{
    const int i = blockIdx.x * blockDim.x + threadIdx.x;
    const float* __restrict__ e = emb + (size_t)i * D_EMB;
    float s = 0.0f;
    #pragma unroll 8
    for (int d = 0; d < D_EMB; ++d) s = fmaf(e[d], e[d], s);
    sq[i] = s;
}

// deg[row] = sum_j A[row,j]  (one block per row, LDS tree reduction)
__global__ void rowsum_f32(const float* __restrict__ Amat,
                           float* __restrict__ deg)
{
    __shared__ float red[256];
    const int row = blockIdx.x;
    const float* __restrict__ a = Amat + (size_t)row * NN;
    float s = 0.0f;
    for (int j = threadIdx.x; j < NN; j += 256) s += a[j];
    red[threadIdx.x] = s;
    __syncthreads();
    for (int off = 128; off > 0; off >>= 1) {
        if (threadIdx.x < off) red[threadIdx.x] += red[threadIdx.x + off];
        __syncthreads();
    }
    if (threadIdx.x == 0) deg[row] = red[0];
}

// ---------------------------------------------------------------------------
// Orchestration
// ---------------------------------------------------------------------------
extern "C" void kernel_launch(void* const* d_in, const int* in_sizes, int n_in,
                              void* d_out, int out_size, void* d_ws, size_t ws_size,
                              hipStream_t stream) {
    (void)in_sizes; (void)n_in; (void)out_size; (void)ws_size;

    const float* x     = (const float*)d_in[0];
    // d_in[1] = adj (only its shape is used by the reference)
    const float* W0    = (const float*)d_in[2];
    const float* b0    = (const float*)d_in[3];
    const float* W1    = (const float*)d_in[4];
    const float* b1    = (const float*)d_in[5];
    const float* W2    = (const float*)d_in[6];
    const float* b2    = (const float*)d_in[7];
    const float* temp  = (const float*)d_in[8];
    const float* theta = (const float*)d_in[9];

    // Workspace carve-up (floats)
    float* ws     = (float*)d_ws;
    float* out_x  = ws;                           // 2048*256
    float* emb    = out_x  + (size_t)NN * HH;     // 2048*64
    float* x_last = emb    + (size_t)NN * D_EMB;  // 2048*128
    float* sq     = x_last + (size_t)NN * D_IN;   // 2048
    float* deg    = sq + NN;                      // 2048

    // Output layout: out [2048*128] then adj_wts [2048*2048]
    float* out_p = (float*)d_out;
    float* adj_p = out_p + (size_t)NN * D_IN;

    const int BLK = 128;            // 4 waves/block, 1 strip/wave
    const int WPB = BLK / 32;

    // 1) out_x = relu(x @ W0 + b0)   [2048 x 256], K=128; strips = 128*4
    wmma_gemm_strip<HH, D_IN, 4, 1><<<(128 * 4) / WPB, BLK, 0, stream>>>(
        x, W0, b0, nullptr, out_x);

    // 2) emb = relu(out_x @ W1 + b1) [2048 x 64], K=256; strips = 128*1
    wmma_gemm_strip<D_EMB, HH, 4, 1><<<(128 * 1) / WPB, BLK, 0, stream>>>(
        out_x, W1, b1, nullptr, emb);

    // 3) x_last = x @ W2 + b2        [2048 x 128], K=128; strips = 128*2
    wmma_gemm_strip<D_IN, D_IN, 4, 0><<<(128 * 2) / WPB, BLK, 0, stream>>>(
        x, W2, b2, nullptr, x_last);

    // 4) sq[i] = ||emb_i||^2
    rowsq_f32<<<NN / 256, 256, 0, stream>>>(emb, sq);

    // 5) adj_wts = sigmoid((1+temp)*dist + (5+theta)) + I  (in-place in d_out,
    //    also the A operand of the final GEMM); strips = 128*32
    gram_prob_strip<4><<<(128 * 32) / WPB, BLK, 0, stream>>>(
        emb, sq, temp, theta, adj_p);

    // 6) deg = rowsum(A)
    rowsum_f32<<<NN, 256, 0, stream>>>(adj_p, deg);

    // 7) out = relu((A @ x_last) / deg)  [2048 x 128], K=2048;
    //    NSTRIP=2 keeps 512 waves in flight for latency hiding.
    wmma_gemm_strip<D_IN, NN, 2, 2><<<(128 * 4) / WPB, BLK, 0, stream>>>(
        adj_p, x_last, nullptr, deg, out_p);
}